// WaveletDomainLoss_79113297592488
// MI455X (gfx1250) — compile-verified
//
#include <hip/hip_runtime.h>

// Haar wavelet L1 loss, exploiting linearity of the DWT:
//   haar(pred) - haar(target) == haar(pred - target)
// Memory-bound streaming kernel: ~201 MB traffic -> ~8.6 us at 23.3 TB/s.
// Reduction uses V_WMMA_F32_16X16X4_F32 (ones-matmul trick) for the wave32 sum.

typedef float v2f __attribute__((ext_vector_type(2)));
typedef float v4f __attribute__((ext_vector_type(4)));
typedef float v8f __attribute__((ext_vector_type(8)));

// Full wave32 sum via one WMMA:
//   A(16x4): lane L<16 -> {VGPR0=A[L][0], VGPR1=A[L][1]}, lane L>=16 -> {A[L-16][2], A[L-16][3]}
//   With VGPR0 = partial, VGPR1 = 0:  A[m][k] = {p[m], 0, p[m+16], 0}
//   B(4x16) = all ones  ->  D[m][n] = p[m] + p[m+16]  (independent of n)
//   D layout: VGPR r, lanes<16 hold D[r][*], lanes>=16 hold D[8+r][*].
//   Sum of 8 D VGPRs: lanes<16 = sum(p[0..7],p[16..23]); lanes>=16 = sum(p[8..15],p[24..31]).
//   One xor-16 shuffle + add -> full 32-lane sum in every lane. EXEC must be all ones.
__device__ __forceinline__ float wave_sum32(float x) {
  v2f a; a.x = x;    a.y = 0.0f;
  v2f b; b.x = 1.0f; b.y = 1.0f;
  v8f c = {};
  v8f d = __builtin_amdgcn_wmma_f32_16x16x4_f32(
      /*neg_a=*/false, a, /*neg_b=*/false, b,
      /*c_mod=*/(short)0, c, /*reuse_a=*/false, /*reuse_b=*/false);
  float s = ((d[0] + d[1]) + (d[2] + d[3])) + ((d[4] + d[5]) + (d[6] + d[7]));
  s += __shfl_xor(s, 16, 32);
  return s;
}

// Each grid-stride iteration handles one "pair": a float4 span of two adjacent
// rows => two 2x2 Haar blocks per tensor. W = 512 floats -> 128 float4 per row,
// row-pair stride = 1024 floats. 4 x b128 non-temporal loads per iteration.
__global__ __launch_bounds__(256) void haar_partial_kernel(
    const float* __restrict__ pred, const float* __restrict__ tgt,
    float* __restrict__ partials, unsigned P) {
  float acc = 0.0f;
  const unsigned stride = gridDim.x * blockDim.x;
  for (unsigned p = blockIdx.x * blockDim.x + threadIdx.x; p < P; p += stride) {
    const unsigned row = p >> 7;                 // / (W/4 = 128)
    const unsigned q   = p & 127u;
    const unsigned top = row * 1024u + q * 4u;   // 2*W floats per row pair
    v4f pr0 = __builtin_nontemporal_load((const v4f*)(pred + top));
    v4f pr1 = __builtin_nontemporal_load((const v4f*)(pred + top + 512u));
    v4f tg0 = __builtin_nontemporal_load((const v4f*)(tgt + top));
    v4f tg1 = __builtin_nontemporal_load((const v4f*)(tgt + top + 512u));
    v4f r0 = pr0 - tg0;   // top row of diff: a0 b0 a1 b1
    v4f r1 = pr1 - tg1;   // bottom row:      c0 d0 c1 d1
    {
      float ab = r0.x + r0.y, cd = r1.x + r1.y;
      float am = r0.x - r0.y, cm = r1.x - r1.y;
      acc += fabsf(ab + cd) + fabsf(ab - cd) + fabsf(am + cm) + fabsf(am - cm);
    }
    {
      float ab = r0.z + r0.w, cd = r1.z + r1.w;
      float am = r0.z - r0.w, cm = r1.z - r1.w;
      acc += fabsf(ab + cd) + fabsf(ab - cd) + fabsf(am + cm) + fabsf(am - cm);
    }
  }
  // Wave reduction (WMMA), then LDS across the 8 waves of this block.
  const float ws = wave_sum32(acc);
  __shared__ float lds[8];
  const unsigned lane = threadIdx.x & 31u;
  const unsigned wid  = threadIdx.x >> 5;
  if (lane == 0) lds[wid] = ws;
  __syncthreads();
  if (threadIdx.x == 0) {
    float t = 0.0f;
#pragma unroll
    for (int i = 0; i < 8; ++i) t += lds[i];
    partials[blockIdx.x] = t;
  }
}

__global__ __launch_bounds__(256) void haar_final_kernel(
    const float* __restrict__ partials, float* __restrict__ out,
    unsigned n, float scale) {
  float s = 0.0f;
  for (unsigned i = threadIdx.x; i < n; i += 256u) s += partials[i];
  const float ws = wave_sum32(s);
  __shared__ float lds[8];
  const unsigned lane = threadIdx.x & 31u;
  const unsigned wid  = threadIdx.x >> 5;
  if (lane == 0) lds[wid] = ws;
  __syncthreads();
  if (threadIdx.x == 0) {
    float t = 0.0f;
#pragma unroll
    for (int i = 0; i < 8; ++i) t += lds[i];
    out[0] = t * scale;   // scale = LOSS_WEIGHT * 0.5 / N
  }
}

extern "C" void kernel_launch(void* const* d_in, const int* in_sizes, int n_in,
                              void* d_out, int out_size, void* d_ws, size_t ws_size,
                              hipStream_t stream) {
  const float* pred = (const float*)d_in[0];
  const float* tgt  = (const float*)d_in[1];
  float* out        = (float*)d_out;
  float* partials   = (float*)d_ws;

  const unsigned N = (unsigned)in_sizes[0];   // 16*6*512*512 = 25,165,824
  const unsigned P = N / 8u;                  // pairs of 2x2 blocks (2 rows x 4 cols)

  int NB = 1024;                              // one float partial per workgroup
  if ((size_t)NB * sizeof(float) > ws_size) NB = (int)(ws_size / sizeof(float));
  if (NB < 1) NB = 1;

  haar_partial_kernel<<<NB, 256, 0, stream>>>(pred, tgt, partials, P);
  haar_final_kernel<<<1, 256, 0, stream>>>(partials, out, (unsigned)NB,
                                           0.5f / (float)N);
}